// NodeAttention_3015067042080
// MI455X (gfx1250) — compile-verified
//
#include <hip/hip_runtime.h>
#include <hip/hip_bf16.h>

#define NN 50000
#define EE 1600000
#define DIMN 64
#define HEADS 8
#define EDIM 32

typedef __attribute__((ext_vector_type(2))) float v2f;
typedef __attribute__((ext_vector_type(8))) float v8f;

__device__ inline v8f wmma_f32_4(v2f a, v2f b, v8f c) {
    // V_WMMA_F32_16X16X4_F32 : 8-arg form (neg_a, A, neg_b, B, c_mod, C, reuse_a, reuse_b)
    return __builtin_amdgcn_wmma_f32_16x16x4_f32(false, a, false, b, (short)0, c, false, false);
}

// monotone float->uint mapping so unsigned atomicMax == float max (handles negatives)
__device__ inline unsigned encF(float f) {
    unsigned u = __float_as_uint(f);
    return (u & 0x80000000u) ? ~u : (u | 0x80000000u);
}
__device__ inline float decF(unsigned e) {
    return (e & 0x80000000u) ? __uint_as_float(e & 0x7FFFFFFFu) : __uint_as_float(~e);
}

// ---------------- init ----------------
__global__ __launch_bounds__(256) void init_kernel(unsigned* __restrict__ smax,
                                                   float* __restrict__ ssum,
                                                   float* __restrict__ attn) {
    int idx = blockIdx.x * blockDim.x + threadIdx.x;
    if (idx < NN * DIMN) attn[idx] = 0.0f;
    if (idx < NN * HEADS) { smax[idx] = 0u; ssum[idx] = 0.0f; }
}

// ---------------- QKV projection: per 16-node block, 4 waves = 4 col tiles ----------------
__global__ __launch_bounds__(128) void qkv_kernel(const float* __restrict__ x,
        const float* __restrict__ WQ, const float* __restrict__ bQ,
        const float* __restrict__ WK, const float* __restrict__ bK,
        const float* __restrict__ WV, const float* __restrict__ bV,
        float* __restrict__ Q, float* __restrict__ K, float* __restrict__ V) {
    const int nodeBase = blockIdx.x * 16;
    const int lane  = threadIdx.x & 31;
    const int ct    = threadIdx.x >> 5;        // col tile 0..3
    const int m     = lane & 15;               // A row / D col within tile
    const int khalf = lane >> 4;               // 0 or 1
    const int col   = ct * 16 + m;

    const float* Ws[3] = {WQ, WK, WV};
    const float* bs[3] = {bQ, bK, bV};
    float*       Os[3] = {Q, K, V};

    #pragma unroll
    for (int mat = 0; mat < 3; ++mat) {
        const float* W = Ws[mat];
        const float bval = bs[mat][col];
        v8f acc;
        #pragma unroll
        for (int r = 0; r < 8; ++r) acc[r] = bval;   // bias depends on col only
        #pragma unroll
        for (int k = 0; k < DIMN; k += 4) {
            const int k0 = k + khalf * 2;
            v2f a, b;
            a.x = x[(size_t)(nodeBase + m) * DIMN + k0];
            a.y = x[(size_t)(nodeBase + m) * DIMN + k0 + 1];
            b.x = W[(size_t)k0 * DIMN + col];
            b.y = W[(size_t)(k0 + 1) * DIMN + col];
            acc = wmma_f32_4(a, b, acc);
        }
        float* O = Os[mat];
        #pragma unroll
        for (int r = 0; r < 8; ++r)
            O[(size_t)(nodeBase + r + khalf * 8) * DIMN + col] = acc[r];
    }
}

// ---------------- edge bias MLP: silu(ea@We1+be1)@We2+be2 ----------------
__global__ __launch_bounds__(128) void edge_mlp_kernel(const float* __restrict__ ea,
        const float* __restrict__ We1, const float* __restrict__ be1,
        const float* __restrict__ We2, const float* __restrict__ be2,
        float* __restrict__ bias) {
    __shared__ float hbuf[4][16][EDIM + 1];   // per-wave staging, padded
    const int lane  = threadIdx.x & 31;
    const int wave  = threadIdx.x >> 5;
    const int eBase = (blockIdx.x * 4 + wave) * 16;
    const int m     = lane & 15;
    const int khalf = lane >> 4;

    // layer 1 via WMMA: 16 edges x 32 hidden, two 16-wide col tiles
    #pragma unroll
    for (int ctile = 0; ctile < 2; ++ctile) {
        const int col = ctile * 16 + m;
        const float bval = be1[col];
        v8f acc;
        #pragma unroll
        for (int r = 0; r < 8; ++r) acc[r] = bval;
        #pragma unroll
        for (int k = 0; k < EDIM; k += 4) {
            const int k0 = k + khalf * 2;
            v2f a, b;
            a.x = ea[(size_t)(eBase + m) * EDIM + k0];
            a.y = ea[(size_t)(eBase + m) * EDIM + k0 + 1];
            b.x = We1[(size_t)k0 * EDIM + col];
            b.y = We1[(size_t)(k0 + 1) * EDIM + col];
            acc = wmma_f32_4(a, b, acc);
        }
        #pragma unroll
        for (int r = 0; r < 8; ++r) {
            float v = acc[r];
            v = v * (1.0f / (1.0f + __expf(-v)));          // SiLU
            hbuf[wave][r + khalf * 8][col] = v;
        }
    }
    __syncthreads();
    // layer 2 (32 -> 8) scalar from LDS: lane handles edge m, 4 heads
    const int hbase = khalf * 4;
    #pragma unroll
    for (int hh = 0; hh < 4; ++hh) {
        const int h = hbase + hh;
        float s = be2[h];
        #pragma unroll
        for (int k = 0; k < EDIM; ++k)
            s += hbuf[wave][m][k] * We2[k * HEADS + h];
        bias[(size_t)(eBase + m) * HEADS + h] = s;
    }
}

// ---------------- per-edge scores + segment max ----------------
__global__ __launch_bounds__(256) void scores_kernel(const int* __restrict__ ei,
        const float* __restrict__ Q, const float* __restrict__ K,
        float* __restrict__ scores /* holds bias on entry, scores on exit */,
        unsigned* __restrict__ smax) {
    const int e = blockIdx.x * blockDim.x + threadIdx.x;
    if (e >= EE) return;
    const int i = ei[e];
    const int j = ei[EE + e];
    const float4* q = (const float4*)(Q + (size_t)j * DIMN);
    const float4* k = (const float4*)(K + (size_t)i * DIMN);
    const float rsq = 0.35355339059327373f;   // 1/sqrt(d_k=8)
    #pragma unroll
    for (int h = 0; h < HEADS; ++h) {
        float4 q0 = q[h * 2], q1 = q[h * 2 + 1];
        float4 k0 = k[h * 2], k1 = k[h * 2 + 1];
        float d = q0.x * k0.x + q0.y * k0.y + q0.z * k0.z + q0.w * k0.w
                + q1.x * k1.x + q1.y * k1.y + q1.z * k1.z + q1.w * k1.w;
        d = d * rsq + scores[(size_t)e * HEADS + h];
        scores[(size_t)e * HEADS + h] = d;
        atomicMax(&smax[j * HEADS + h], encF(d));
    }
}

// ---------------- exp(s - max) + segment sum ----------------
__global__ __launch_bounds__(256) void expsum_kernel(const int* __restrict__ ei,
        float* __restrict__ scores, const unsigned* __restrict__ smax,
        float* __restrict__ ssum) {
    const long long idx = (long long)blockIdx.x * blockDim.x + threadIdx.x;
    if (idx >= (long long)EE * HEADS) return;
    const int e = (int)(idx >> 3);
    const int h = (int)(idx & 7);
    const int j = ei[EE + e];
    const float m = decF(smax[j * HEADS + h]);
    const float v = __expf(scores[idx] - m);
    scores[idx] = v;
    atomicAdd(&ssum[j * HEADS + h], v);
}

// ---------------- alpha * V scatter ----------------
__global__ __launch_bounds__(256) void scatter_kernel(const int* __restrict__ ei,
        const float* __restrict__ scores, const float* __restrict__ ssum,
        const float* __restrict__ V, float* __restrict__ attn) {
    const long long idx = (long long)blockIdx.x * blockDim.x + threadIdx.x;
    if (idx >= (long long)EE * DIMN) return;
    const int e = (int)(idx >> 6);
    const int d = (int)(idx & 63);
    const int h = d >> 3;
    const int i = ei[e];
    const int j = ei[EE + e];
    const float alpha = scores[(size_t)e * HEADS + h] / (ssum[j * HEADS + h] + 1e-12f);
    atomicAdd(&attn[(size_t)j * DIMN + d], alpha * V[(size_t)i * DIMN + d]);
}

// ---------------- output projection ----------------
__global__ __launch_bounds__(128) void outproj_kernel(const float* __restrict__ attn,
        const float* __restrict__ WO, const float* __restrict__ bO,
        float* __restrict__ out) {
    const int nodeBase = blockIdx.x * 16;
    const int lane  = threadIdx.x & 31;
    const int ct    = threadIdx.x >> 5;
    const int m     = lane & 15;
    const int khalf = lane >> 4;
    const int col   = ct * 16 + m;
    const float bval = bO[col];
    v8f acc;
    #pragma unroll
    for (int r = 0; r < 8; ++r) acc[r] = bval;
    #pragma unroll
    for (int k = 0; k < DIMN; k += 4) {
        const int k0 = k + khalf * 2;
        v2f a, b;
        a.x = attn[(size_t)(nodeBase + m) * DIMN + k0];
        a.y = attn[(size_t)(nodeBase + m) * DIMN + k0 + 1];
        b.x = WO[(size_t)k0 * DIMN + col];
        b.y = WO[(size_t)(k0 + 1) * DIMN + col];
        acc = wmma_f32_4(a, b, acc);
    }
    #pragma unroll
    for (int r = 0; r < 8; ++r)
        out[(size_t)(nodeBase + r + khalf * 8) * DIMN + col] = acc[r];
}

extern "C" void kernel_launch(void* const* d_in, const int* in_sizes, int n_in,
                              void* d_out, int out_size, void* d_ws, size_t ws_size,
                              hipStream_t stream) {
    const float* x   = (const float*)d_in[0];
    const int*   ei  = (const int*)  d_in[1];
    const float* ea  = (const float*)d_in[2];
    const float* WQ  = (const float*)d_in[3];
    const float* bQ  = (const float*)d_in[4];
    const float* WK  = (const float*)d_in[5];
    const float* bK  = (const float*)d_in[6];
    const float* WV  = (const float*)d_in[7];
    const float* bV  = (const float*)d_in[8];
    const float* WO  = (const float*)d_in[9];
    const float* bO  = (const float*)d_in[10];
    const float* We1 = (const float*)d_in[11];
    const float* be1 = (const float*)d_in[12];
    const float* We2 = (const float*)d_in[13];
    const float* be2 = (const float*)d_in[14];
    float* out = (float*)d_out;

    // workspace carve-up (floats)
    float* w = (float*)d_ws;
    const size_t ND = (size_t)NN * DIMN;          // 3.2M
    float*    Q      = w;
    float*    K      = w + ND;
    float*    V      = w + 2 * ND;
    float*    scores = w + 3 * ND;                 // [E,8], bias then scores then exp-vals
    unsigned* smax   = (unsigned*)(scores + (size_t)EE * HEADS);
    float*    ssum   = (float*)smax + (size_t)NN * HEADS;
    float*    attn   = ssum + (size_t)NN * HEADS;

    init_kernel<<<(NN * DIMN + 255) / 256, 256, 0, stream>>>(smax, ssum, attn);

    qkv_kernel<<<NN / 16, 128, 0, stream>>>(x, WQ, bQ, WK, bK, WV, bV, Q, K, V);

    edge_mlp_kernel<<<EE / 64, 128, 0, stream>>>(ea, We1, be1, We2, be2, scores);

    scores_kernel<<<(EE + 255) / 256, 256, 0, stream>>>(ei, Q, K, scores, smax);

    {
        long long n = (long long)EE * HEADS;
        expsum_kernel<<<(unsigned)((n + 255) / 256), 256, 0, stream>>>(ei, scores, smax, ssum);
    }
    {
        long long n = (long long)EE * DIMN;
        scatter_kernel<<<(unsigned)((n + 255) / 256), 256, 0, stream>>>(ei, scores, ssum, V, attn);
    }

    outproj_kernel<<<NN / 16, 128, 0, stream>>>(attn, WO, bO, out);
}